// SplineconvModel_79121887527328
// MI455X (gfx1250) — compile-verified
//
#include <hip/hip_runtime.h>

typedef float v2f __attribute__((ext_vector_type(2)));
typedef float v8f __attribute__((ext_vector_type(8)));

#define N_NODES 100000
#define N_EDGES 1000000
#define CC      64
#define KK      5
#define KCAT    (KK * CC + CC)   /* 384 : [S | x] concatenated K dimension */
#define ASTRIDE (KCAT + 4)       /* 388 : pad to kill LDS bank conflicts   */

// ---------------------------------------------------------------------------
// CSR build (once per launch; dst is layer-invariant)
// ---------------------------------------------------------------------------
__global__ void hist_kernel(const int* __restrict__ dst, int* __restrict__ cnt) {
    int e = blockIdx.x * 256 + threadIdx.x;
    if (e < N_EDGES) atomicAdd(&cnt[dst[e]], 1);
}

// Single-block exclusive scan over cnt[N] -> rowstart[N+1], cursor copy.
__global__ void __launch_bounds__(1024)
scan_kernel(const int* __restrict__ cnt, int* __restrict__ rowstart,
            int* __restrict__ cursor) {
    __shared__ int smem[1024];
    __shared__ int carry_s;
    const int tid = threadIdx.x;
    if (tid == 0) carry_s = 0;
    __syncthreads();
    for (int base = 0; base < N_NODES; base += 1024) {
        int v = (base + tid < N_NODES) ? cnt[base + tid] : 0;
        smem[tid] = v;
        __syncthreads();
        for (int off = 1; off < 1024; off <<= 1) {
            int t = (tid >= off) ? smem[tid - off] : 0;
            __syncthreads();
            smem[tid] += t;
            __syncthreads();
        }
        int excl = carry_s + smem[tid] - v;
        if (base + tid < N_NODES) { rowstart[base + tid] = excl; cursor[base + tid] = excl; }
        __syncthreads();
        if (tid == 1023) carry_s += smem[1023];
        __syncthreads();
    }
    if (tid == 0) rowstart[N_NODES] = carry_s;   // == N_EDGES
}

__global__ void invdeg_kernel(const int* __restrict__ cnt, float* __restrict__ invdeg) {
    int n = blockIdx.x * 256 + threadIdx.x;
    if (n < N_NODES) invdeg[n] = 1.0f / fmaxf((float)cnt[n], 1.0f);
}

// Sort edge records into dst-order; fold 1/deg into the spline weights.
__global__ void fill_kernel(const int* __restrict__ src, const int* __restrict__ dst,
                            const float* __restrict__ u, const float* __restrict__ invdeg,
                            int* __restrict__ cursor,
                            int* __restrict__ esrc, int* __restrict__ epack,
                            float* __restrict__ ew0, float* __restrict__ ew1) {
    int e = blockIdx.x * 256 + threadIdx.x;
    if (e >= N_EDGES) return;
    int d = dst[e];
    float uu = fminf(fmaxf(u[e], 0.0f), 1.0f - 1e-6f);
    float p  = uu * (float)(KK - 1);
    int   i0 = (int)p;                  // floor, p >= 0, i0 in [0,3]
    float fr = p - (float)i0;
    float iv = invdeg[d];
    int pos = atomicAdd(&cursor[d], 1);
    esrc[pos]  = src[e];
    epack[pos] = (d << 3) | i0;
    ew0[pos]   = (1.0f - fr) * iv;
    ew1[pos]   = fr * iv;
}

// ---------------------------------------------------------------------------
// Fused layer: per block = 16 dst rows.
//   1) stage x rows (root term) + zero spline slots in LDS A-panel
//   2) aggregate incident edges into A-panel via ds_add_f32 (no global atomics)
//   3) 16x384 x 384x64 WMMA contraction (v_wmma_f32_16x16x4_f32), bias+ReLU
// ---------------------------------------------------------------------------
__global__ void __launch_bounds__(128)
spline_layer_kernel(const float* __restrict__ x,
                    const int*   __restrict__ rowstart,
                    const int*   __restrict__ esrc,
                    const int*   __restrict__ epack,
                    const float* __restrict__ ew0,
                    const float* __restrict__ ew1,
                    const float* __restrict__ W,    // [K,Cin,Cout] flat
                    const float* __restrict__ R,    // [Cin,Cout]
                    const float* __restrict__ bias,
                    float*       __restrict__ out) {
    __shared__ float As[16 * ASTRIDE];     // ~24.8 KB
    __shared__ float Bs[KCAT * CC];        // 96 KB (total ~121 KB < 320 KB WGP LDS)

    const int r0   = blockIdx.x * 16;
    const int tid  = threadIdx.x;
    const int wave = tid >> 5;
    const int lane = tid & 31;

    // Zero the spline part of A; stage x rows into the last 64 columns (float4).
    for (int i = tid; i < 16 * (KK * CC); i += 128) {
        int m = i / (KK * CC);
        int c = i - m * (KK * CC);
        As[m * ASTRIDE + c] = 0.0f;
    }
    for (int i = tid; i < 16 * (CC / 4); i += 128) {
        int m = i >> 4, c4 = i & 15;
        float4 v = ((const float4*)x)[(size_t)(r0 + m) * (CC / 4) + c4];
        float* p = &As[m * ASTRIDE + KK * CC + c4 * 4];
        p[0] = v.x; p[1] = v.y; p[2] = v.z; p[3] = v.w;
    }
    // Stage B = [W ; R] (float4 loads; tiny, L0/L2 hot).
    for (int i = tid; i < KCAT * (CC / 4); i += 128) {
        int k = i >> 4, o4 = i & 15;
        float4 v = (k < KK * CC) ? ((const float4*)W)[(size_t)k * (CC / 4) + o4]
                                 : ((const float4*)R)[(size_t)(k - KK * CC) * (CC / 4) + o4];
        float* p = &Bs[k * CC + o4 * 4];
        p[0] = v.x; p[1] = v.y; p[2] = v.z; p[3] = v.w;
    }
    __syncthreads();

    // Edge aggregation: one wave per edge (round-robin), 2 channels per lane.
    const int ebeg = rowstart[r0];
    const int eend = rowstart[r0 + 16];
    for (int e = ebeg + wave; e < eend; e += 4) {
        int   s  = esrc[e];
        int   pk = epack[e];
        int   m  = (pk >> 3) - r0;
        int   k0 = pk & 7;
        float w0 = ew0[e];
        float w1 = ew1[e];
        float v0 = x[(size_t)s * CC + lane];        // coalesced 128B
        float v1 = x[(size_t)s * CC + lane + 32];
        float* rowp = &As[m * ASTRIDE + k0 * CC];
        atomicAdd(&rowp[lane],           w0 * v0);  // ds_add_f32
        atomicAdd(&rowp[lane + 32],      w0 * v1);
        atomicAdd(&rowp[CC + lane],      w1 * v0);
        atomicAdd(&rowp[CC + lane + 32], w1 * v1);
    }
    __syncthreads();

    // WMMA contraction: 4 waves x one 16x16 N-tile, 96 chained K=4 steps.
    const int half = lane >> 4;
    const int row  = lane & 15;
    const int o0   = wave * 16;

    v8f acc = {0.f, 0.f, 0.f, 0.f, 0.f, 0.f, 0.f, 0.f};
#pragma unroll 4
    for (int k0 = 0; k0 < KCAT; k0 += 4) {
        int ka = k0 + 2 * half;
        v2f a = { As[row * ASTRIDE + ka], As[row * ASTRIDE + ka + 1] };
        v2f b = { Bs[ka * CC + o0 + row], Bs[(ka + 1) * CC + o0 + row] };
        acc = __builtin_amdgcn_wmma_f32_16x16x4_f32(
                  false, a, false, b, (short)0, acc, false, false);
    }

    // D layout: element v -> M = v + 8*half, N = lane&15. Fuse bias + ReLU.
    const int o = o0 + row;
    const float bo = bias[o];
#pragma unroll
    for (int v = 0; v < 8; ++v) {
        int m = r0 + v + 8 * half;
        out[(size_t)m * CC + o] = fmaxf(acc[v] + bo, 0.0f);
    }
}

// ---------------------------------------------------------------------------
extern "C" void kernel_launch(void* const* d_in, const int* in_sizes, int n_in,
                              void* d_out, int out_size, void* d_ws, size_t ws_size,
                              hipStream_t stream) {
    (void)in_sizes; (void)n_in; (void)out_size; (void)ws_size;

    const float* x0  = (const float*)d_in[0];
    const int*   ei  = (const int*)d_in[1];
    const float* ea  = (const float*)d_in[2];
    const int*   src = ei;
    const int*   dst = ei + N_EDGES;

    // Workspace carve-up (~68 MB)
    char* ws = (char*)d_ws;
    int*   esrc     = (int*)ws;   ws += (size_t)N_EDGES * sizeof(int);
    int*   epack    = (int*)ws;   ws += (size_t)N_EDGES * sizeof(int);
    float* ew0      = (float*)ws; ws += (size_t)N_EDGES * sizeof(float);
    float* ew1      = (float*)ws; ws += (size_t)N_EDGES * sizeof(float);
    int*   cnt      = (int*)ws;   ws += (size_t)N_NODES * sizeof(int);
    int*   rowstart = (int*)ws;   ws += (size_t)(N_NODES + 1) * sizeof(int);
    int*   cursor   = (int*)ws;   ws += (size_t)N_NODES * sizeof(int);
    float* invdeg   = (float*)ws; ws += (size_t)N_NODES * sizeof(float);
    float* xA       = (float*)ws; ws += (size_t)N_NODES * CC * sizeof(float);
    float* xB       = (float*)ws;

    // CSR by destination, built once; 1/deg folded into edge weights.
    hipMemsetAsync(cnt, 0, (size_t)N_NODES * sizeof(int), stream);
    hist_kernel<<<(N_EDGES + 255) / 256, 256, 0, stream>>>(dst, cnt);
    scan_kernel<<<1, 1024, 0, stream>>>(cnt, rowstart, cursor);
    invdeg_kernel<<<(N_NODES + 255) / 256, 256, 0, stream>>>(cnt, invdeg);
    fill_kernel<<<(N_EDGES + 255) / 256, 256, 0, stream>>>(
        src, dst, ea, invdeg, cursor, esrc, epack, ew0, ew1);

    const float* xin = x0;
    float* outs[3] = { xA, xB, (float*)d_out };
    for (int l = 0; l < 3; ++l) {
        const float* W = (const float*)d_in[3 + 3 * l];
        const float* R = (const float*)d_in[4 + 3 * l];
        const float* b = (const float*)d_in[5 + 3 * l];
        spline_layer_kernel<<<N_NODES / 16, 128, 0, stream>>>(
            xin, rowstart, esrc, epack, ew0, ew1, W, R, b, outs[l]);
        xin = outs[l];
    }
}